// HeteroRGCNLayer_8624294330466
// MI455X (gfx1250) — compile-verified
//
#include <hip/hip_runtime.h>
#include <hip/hip_bf16.h>

typedef __attribute__((ext_vector_type(2))) float v2f;
typedef __attribute__((ext_vector_type(8))) float v8f;

#define DIM 512

// ---------------------------------------------------------------------------
// Zero-fill (float4 granularity)
// ---------------------------------------------------------------------------
__global__ void zero_f4(float* __restrict__ p, long long n) {
    long long i = ((long long)blockIdx.x * blockDim.x + threadIdx.x) * 4;
    if (i < n) {
        float4 z = {0.f, 0.f, 0.f, 0.f};
        *(float4*)(p + i) = z;
    }
}

// ---------------------------------------------------------------------------
// Transpose 512x512: WT[n][k] = W[k][n]
// ---------------------------------------------------------------------------
__global__ void transpose512(const float* __restrict__ W, float* __restrict__ WT) {
    int i = blockIdx.x * blockDim.x + threadIdx.x;   // i = k*512 + n
    int k = i >> 9;
    int n = i & 511;
    WT[n * DIM + k] = W[i];
}

// ---------------------------------------------------------------------------
// GEMM: C[M x 512] = A[M x 512] @ W[512 x 512] + bias, via WMMA f32 16x16x4.
// WT is W transposed (N-major) so B-fragment K-pairs are contiguous.
// Block = 256 threads = 8 waves; block owns a 32-row M stripe.
// Each wave owns a 32x64 output tile: 2 M-tiles x 4 N-tiles = 8 accumulators.
// Per K-step (depth 4): 2 A-fragment loads + 4 B-fragment loads -> 8 WMMAs.
// ---------------------------------------------------------------------------
__global__ __launch_bounds__(256)
void gemm_wmma_f32(const float* __restrict__ A, const float* __restrict__ WT,
                   const float* __restrict__ bias, float* __restrict__ C, int M) {
    const int wave = threadIdx.x >> 5;
    const int lane = threadIdx.x & 31;
    const int half = lane >> 4;          // 0: K+0..1, 1: K+2..3
    const int l16  = lane & 15;
    const int mBase = blockIdx.x * 32;

    // A-fragment rows (clamped; no divergence before WMMA)
    int m0 = mBase + l16;        if (m0 >= M) m0 = M - 1;
    int m1 = mBase + 16 + l16;   if (m1 >= M) m1 = M - 1;

    const int n0 = wave * 64;
    const int kOff = half * 2;

    const float* Arow0 = A + (size_t)m0 * DIM + kOff;
    const float* Arow1 = A + (size_t)m1 * DIM + kOff;
    const float* Bcol0 = WT + (size_t)(n0 + l16) * DIM + kOff;

    v8f acc[2][4] = {};

    for (int k0 = 0; k0 < DIM; k0 += 4) {
        v2f a0 = *(const v2f*)(Arow0 + k0);
        v2f a1 = *(const v2f*)(Arow1 + k0);
        v2f b0 = *(const v2f*)(Bcol0 + k0 + 0 * 16 * DIM);
        v2f b1 = *(const v2f*)(Bcol0 + k0 + 1 * 16 * DIM);
        v2f b2 = *(const v2f*)(Bcol0 + k0 + 2 * 16 * DIM);
        v2f b3 = *(const v2f*)(Bcol0 + k0 + 3 * 16 * DIM);
        acc[0][0] = __builtin_amdgcn_wmma_f32_16x16x4_f32(false, a0, false, b0, (short)0, acc[0][0], false, false);
        acc[0][1] = __builtin_amdgcn_wmma_f32_16x16x4_f32(false, a0, false, b1, (short)0, acc[0][1], false, false);
        acc[0][2] = __builtin_amdgcn_wmma_f32_16x16x4_f32(false, a0, false, b2, (short)0, acc[0][2], false, false);
        acc[0][3] = __builtin_amdgcn_wmma_f32_16x16x4_f32(false, a0, false, b3, (short)0, acc[0][3], false, false);
        acc[1][0] = __builtin_amdgcn_wmma_f32_16x16x4_f32(false, a1, false, b0, (short)0, acc[1][0], false, false);
        acc[1][1] = __builtin_amdgcn_wmma_f32_16x16x4_f32(false, a1, false, b1, (short)0, acc[1][1], false, false);
        acc[1][2] = __builtin_amdgcn_wmma_f32_16x16x4_f32(false, a1, false, b2, (short)0, acc[1][2], false, false);
        acc[1][3] = __builtin_amdgcn_wmma_f32_16x16x4_f32(false, a1, false, b3, (short)0, acc[1][3], false, false);
    }

    // C/D layout: VGPR r -> rows tileBase+r (lanes 0-15) and tileBase+r+8 (lanes 16-31)
#pragma unroll
    for (int mt = 0; mt < 2; ++mt) {
        const int rowBase = mBase + mt * 16 + half * 8;
#pragma unroll
        for (int t = 0; t < 4; ++t) {
            const int n = n0 + t * 16 + l16;
            const float bs = bias[n];
#pragma unroll
            for (int r = 0; r < 8; ++r) {
                const int row = rowBase + r;
                if (row < M) C[(size_t)row * DIM + n] = acc[mt][t][r] + bs;
            }
        }
    }
}

// ---------------------------------------------------------------------------
// In-degree counts
// ---------------------------------------------------------------------------
__global__ void count_edges(const int* __restrict__ dst, float* __restrict__ cnt, int E) {
    int e = blockIdx.x * blockDim.x + threadIdx.x;
    if (e < E) atomicAdd(&cnt[dst[e]], 1.0f);
}

// ---------------------------------------------------------------------------
// Scatter-add messages: 128 threads per edge, 4 floats per thread.
// ---------------------------------------------------------------------------
__global__ void scatter_add(const float* __restrict__ Wh, const int* __restrict__ src,
                            const int* __restrict__ dst, float* __restrict__ out, int E) {
    long long gid = (long long)blockIdx.x * blockDim.x + threadIdx.x;
    int e = (int)(gid >> 7);
    if (e >= E) return;
    int c = ((int)gid & 127) << 2;
    int s = src[e], d = dst[e];
    float4 v = *(const float4*)(Wh + (size_t)s * DIM + c);
    float* o = out + (size_t)d * DIM + c;
    atomicAdd(o + 0, v.x);
    atomicAdd(o + 1, v.y);
    atomicAdd(o + 2, v.z);
    atomicAdd(o + 3, v.w);
}

// ---------------------------------------------------------------------------
// buf[i] /= max(cnt[row],1)      (4 floats per thread, same row)
// ---------------------------------------------------------------------------
__global__ void scale_rows(float* __restrict__ buf, const float* __restrict__ cnt, long long n) {
    long long i = ((long long)blockIdx.x * blockDim.x + threadIdx.x) * 4;
    if (i >= n) return;
    int row = (int)(i >> 9);
    float inv = 1.0f / fmaxf(cnt[row], 1.0f);
    float4 v = *(float4*)(buf + i);
    v.x *= inv; v.y *= inv; v.z *= inv; v.w *= inv;
    *(float4*)(buf + i) = v;
}

// ---------------------------------------------------------------------------
// dstb[i] += srcb[i] / max(cnt[row],1)
// ---------------------------------------------------------------------------
__global__ void add_scaled(float* __restrict__ dstb, const float* __restrict__ srcb,
                           const float* __restrict__ cnt, long long n) {
    long long i = ((long long)blockIdx.x * blockDim.x + threadIdx.x) * 4;
    if (i >= n) return;
    int row = (int)(i >> 9);
    float inv = 1.0f / fmaxf(cnt[row], 1.0f);
    float4 s = *(const float4*)(srcb + i);
    float4 d = *(float4*)(dstb + i);
    d.x += s.x * inv; d.y += s.y * inv; d.z += s.z * inv; d.w += s.w * inv;
    *(float4*)(dstb + i) = d;
}

// ---------------------------------------------------------------------------
extern "C" void kernel_launch(void* const* d_in, const int* in_sizes, int n_in,
                              void* d_out, int out_size, void* d_ws, size_t ws_size,
                              hipStream_t stream) {
    const float* feat_A = (const float*)d_in[0];
    const float* feat_B = (const float*)d_in[1];
    const float* W_aa   = (const float*)d_in[2];
    const float* b_aa   = (const float*)d_in[3];
    const float* W_ab   = (const float*)d_in[4];
    const float* b_ab   = (const float*)d_in[5];
    const float* W_ba   = (const float*)d_in[6];
    const float* b_ba   = (const float*)d_in[7];
    const int* src_aa = (const int*)d_in[8];
    const int* dst_aa = (const int*)d_in[9];
    const int* src_ab = (const int*)d_in[10];
    const int* dst_ab = (const int*)d_in[11];
    const int* src_ba = (const int*)d_in[12];
    const int* dst_ba = (const int*)d_in[13];

    const int NA = in_sizes[0] / DIM;
    const int NB = in_sizes[1] / DIM;
    const int E_aa = in_sizes[8];
    const int E_ab = in_sizes[10];
    const int E_ba = in_sizes[12];

    float* hA = (float*)d_out;
    float* hB = hA + (size_t)NA * DIM;

    float* ws  = (float*)d_ws;
    float* Wh  = ws;                                   // NA*DIM floats (NA==NB)
    float* S   = Wh + (size_t)NA * DIM;                // NA*DIM floats
    float* WT  = S + (size_t)NA * DIM;                 // DIM*DIM floats
    float* cnt = WT + (size_t)DIM * DIM;               // max(NA,NB) floats

    const long long nOut = (long long)(NA + NB) * DIM;
    const long long nA   = (long long)NA * DIM;
    const long long nB   = (long long)NB * DIM;

    dim3 blk(256);
    auto g4 = [](long long n) { return dim3((unsigned)((n / 4 + 255) / 256)); };
    dim3 gT((DIM * DIM) / 256);
    auto gE128 = [](int E) { return dim3((unsigned)(((long long)E * 128 + 255) / 256)); };
    auto gE    = [](int E) { return dim3((unsigned)((E + 255) / 256)); };
    auto gCnt  = [](int N) { return dim3((unsigned)((N / 4 + 255) / 256)); };
    auto gM    = [](int M) { return dim3((unsigned)((M + 31) / 32)); };

    // zero output accumulators
    hipLaunchKernelGGL(zero_f4, g4(nOut), blk, 0, stream, hA, nOut);

    // ---- etype A->A : accumulate into hA, normalize in place ----
    hipLaunchKernelGGL(transpose512, gT, blk, 0, stream, W_aa, WT);
    hipLaunchKernelGGL(gemm_wmma_f32, gM(NA), blk, 0, stream,
                       feat_A, WT, b_aa, Wh, NA);
    hipLaunchKernelGGL(zero_f4, gCnt(NA), blk, 0, stream, cnt, (long long)NA);
    hipLaunchKernelGGL(count_edges, gE(E_aa), blk, 0, stream, dst_aa, cnt, E_aa);
    hipLaunchKernelGGL(scatter_add, gE128(E_aa), blk, 0, stream, Wh, src_aa, dst_aa, hA, E_aa);
    hipLaunchKernelGGL(scale_rows, g4(nA), blk, 0, stream, hA, cnt, nA);

    // ---- etype B->A : accumulate into scratch S, add normalized into hA ----
    hipLaunchKernelGGL(transpose512, gT, blk, 0, stream, W_ba, WT);
    hipLaunchKernelGGL(gemm_wmma_f32, gM(NB), blk, 0, stream,
                       feat_B, WT, b_ba, Wh, NB);
    hipLaunchKernelGGL(zero_f4, g4(nA), blk, 0, stream, S, nA);
    hipLaunchKernelGGL(zero_f4, gCnt(NA), blk, 0, stream, cnt, (long long)NA);
    hipLaunchKernelGGL(count_edges, gE(E_ba), blk, 0, stream, dst_ba, cnt, E_ba);
    hipLaunchKernelGGL(scatter_add, gE128(E_ba), blk, 0, stream, Wh, src_ba, dst_ba, S, E_ba);
    hipLaunchKernelGGL(add_scaled, g4(nA), blk, 0, stream, hA, S, cnt, nA);

    // ---- etype A->B : accumulate into hB, normalize in place ----
    hipLaunchKernelGGL(transpose512, gT, blk, 0, stream, W_ab, WT);
    hipLaunchKernelGGL(gemm_wmma_f32, gM(NA), blk, 0, stream,
                       feat_A, WT, b_ab, Wh, NA);
    hipLaunchKernelGGL(zero_f4, gCnt(NB), blk, 0, stream, cnt, (long long)NB);
    hipLaunchKernelGGL(count_edges, gE(E_ab), blk, 0, stream, dst_ab, cnt, E_ab);
    hipLaunchKernelGGL(scatter_add, gE128(E_ab), blk, 0, stream, Wh, src_ab, dst_ab, hB, E_ab);
    hipLaunchKernelGGL(scale_rows, g4(nB), blk, 0, stream, hB, cnt, nB);
}